// TorchLMHeadCE_11922829213914
// MI455X (gfx1250) — compile-verified
//
#include <hip/hip_runtime.h>
#include <math.h>

// ---- problem constants (match reference) ----
constexpr int N_ROWS = 8192;
constexpr int H_DIM  = 4096;
constexpr int V_DIM  = 32000;
constexpr int IGNORE_INDEX = -100;

// ---- tiling ----
constexpr int BM = 64;     // rows per workgroup
constexpr int BN = 128;    // logit cols per tile
constexpr int BK = 32;     // K-chunk staged in LDS
constexpr int VPB = 5;     // v-tiles per workgroup (BN*VPB = 640 cols)
constexpr int VBLOCKS = V_DIM / (BN * VPB);   // 50
constexpr int LDA = 36;    // padded LDS strides (16B-aligned rows)
constexpr int LDB = 36;
constexpr int LDC = 132;
constexpr int NKC = H_DIM / BK;               // 128 K-chunks

typedef __attribute__((ext_vector_type(2))) float v2f;
typedef __attribute__((ext_vector_type(8))) float v8f;
typedef __attribute__((ext_vector_type(4))) int   v4i;

// gfx1250 async global->LDS path (ASYNCcnt-tracked), with sync fallback
#if defined(__gfx1250__) && \
    __has_builtin(__builtin_amdgcn_global_load_async_to_lds_b128) && \
    __has_builtin(__builtin_amdgcn_s_wait_asynccnt)
#define USE_ASYNC_LDS 1
#else
#define USE_ASYNC_LDS 0
#endif

#if USE_ASYNC_LDS
typedef __attribute__((address_space(1))) v4i* gvec_ptr;   // global int4*
typedef __attribute__((address_space(3))) v4i* lvec_ptr;   // LDS int4*
__device__ __forceinline__ void cp16_async(const float* g, float* l) {
    __builtin_amdgcn_global_load_async_to_lds_b128(
        (gvec_ptr)g, (lvec_ptr)l, /*offset=*/0, /*cpol=*/0);
}
#endif

// -------------------------------------------------------------------------
// Kernel 1: fused GEMM (fp32 WMMA) + online softmax partials + target logit
// grid = (N/BM = 128, VBLOCKS = 50), block = 256 (8 waves: 2(M) x 4(N))
// -------------------------------------------------------------------------
__global__ __launch_bounds__(256)
void ce_partial_kernel(const float* __restrict__ x,
                       const float* __restrict__ W,
                       const int*   __restrict__ target,
                       float* __restrict__ pm,   // [VBLOCKS][N] running max
                       float* __restrict__ ps,   // [VBLOCKS][N] running sumexp
                       float* __restrict__ tgt)  // [N] target logit
{
    __shared__ float Alds[2][BM * LDA];
    __shared__ float Blds[2][BN * LDB];
    __shared__ float Clds[BM * LDC];
    __shared__ float pMax[4][BM];
    __shared__ float pSum[4][BM];

    const int t    = threadIdx.x;
    const int wave = t >> 5;          // 0..7
    const int lane = t & 31;
    const int wm   = wave >> 2;       // 0..1 : 32-row slice
    const int wn   = wave & 3;        // 0..3 : 32-col slice
    const int half = lane >> 4;       // 0/1
    const int lid  = lane & 15;

    const int nBase      = blockIdx.x * BM;
    const int vGroupBase = blockIdx.y * (BN * VPB);

    // staging coordinates: one 16B chunk per (thread, p)
    const int ldRow = t >> 3;         // 0..31
    const int ldCol = (t & 7) * 4;    // 0,4,...,28

    // per-row running softmax stats (threads 0..63 own row nBase+t)
    float m_run = -INFINITY;
    float s_run = 0.0f;

    // stage one BK-chunk of A and B into LDS buffer `buf`
    auto stage = [&](int buf, int k0, int vBase) {
        #pragma unroll
        for (int p = 0; p < 2; ++p) {
            const int row  = ldRow + 32 * p;
            const float* g = x + (size_t)(nBase + row) * H_DIM + k0 + ldCol;
            float* l       = &Alds[buf][row * LDA + ldCol];
#if USE_ASYNC_LDS
            cp16_async(g, l);
#else
            *reinterpret_cast<float4*>(l) = *reinterpret_cast<const float4*>(g);
#endif
        }
        #pragma unroll
        for (int p = 0; p < 4; ++p) {
            const int row  = ldRow + 32 * p;
            const float* g = W + (size_t)(vBase + row) * H_DIM + k0 + ldCol;
            float* l       = &Blds[buf][row * LDB + ldCol];
#if USE_ASYNC_LDS
            cp16_async(g, l);
#else
            *reinterpret_cast<float4*>(l) = *reinterpret_cast<const float4*>(g);
#endif
        }
    };

    for (int vt = 0; vt < VPB; ++vt) {
        const int vBase = vGroupBase + vt * BN;

        v8f acc[2][2];
        #pragma unroll
        for (int i = 0; i < 2; ++i)
            #pragma unroll
            for (int j = 0; j < 2; ++j)
                #pragma unroll
                for (int r = 0; r < 8; ++r)
                    acc[i][j][r] = 0.0f;

        stage(0, 0, vBase);   // prologue: fill buffer 0

        for (int kc = 0; kc < NKC; ++kc) {
            const int cur = kc & 1;

            // issue next chunk into the other buffer (its readers finished at
            // the trailing barrier of iteration kc-1), then wait for `cur`.
            if (kc + 1 < NKC) {
                stage(cur ^ 1, (kc + 1) * BK, vBase);
#if USE_ASYNC_LDS
                __builtin_amdgcn_s_wait_asynccnt(6);   // keep next 6 in flight
#endif
            } else {
#if USE_ASYNC_LDS
                __builtin_amdgcn_s_wait_asynccnt(0);
#endif
            }
            __syncthreads();   // cur buffer visible to all waves

            const float* Ab = Alds[cur];
            const float* Bb = Blds[cur];
            #pragma unroll
            for (int kk = 0; kk < BK; kk += 4) {
                const int kcol = kk + half * 2;
                v2f aF[2], bF[2];
                #pragma unroll
                for (int i = 0; i < 2; ++i) {
                    const int row = wm * 32 + i * 16 + lid;
                    aF[i].x = Ab[row * LDA + kcol];
                    aF[i].y = Ab[row * LDA + kcol + 1];
                }
                #pragma unroll
                for (int j = 0; j < 2; ++j) {
                    const int col = wn * 32 + j * 16 + lid;
                    bF[j].x = Bb[col * LDB + kcol];
                    bF[j].y = Bb[col * LDB + kcol + 1];
                }
                #pragma unroll
                for (int i = 0; i < 2; ++i)
                    #pragma unroll
                    for (int j = 0; j < 2; ++j)
                        acc[i][j] = __builtin_amdgcn_wmma_f32_16x16x4_f32(
                            false, aF[i], false, bF[j],
                            (short)0, acc[i][j], false, false);
            }
            __syncthreads();   // readers of cur done before it is restaged
        }

        // spill accumulators to LDS logits tile [BM][BN]
        // C layout: VGPR r -> M = r + half*8, N = lid (within 16x16 tile)
        #pragma unroll
        for (int i = 0; i < 2; ++i)
            #pragma unroll
            for (int j = 0; j < 2; ++j) {
                const int colL = wn * 32 + j * 16 + lid;
                #pragma unroll
                for (int r = 0; r < 8; ++r) {
                    const int rowL = wm * 32 + i * 16 + r + half * 8;
                    Clds[rowL * LDC + colL] = acc[i][j][r];
                }
            }
        __syncthreads();

        // parallel tile softmax stats: 4 threads per row, 32 cols each
        {
            const int row = t & 63;
            const int seg = t >> 6;            // 0..3
            const float* c = &Clds[row * LDC + seg * 32];
            float mt = -INFINITY;
            #pragma unroll 8
            for (int i = 0; i < 32; ++i) mt = fmaxf(mt, c[i]);
            float st = 0.0f;
            #pragma unroll 8
            for (int i = 0; i < 32; ++i) st += __expf(c[i] - mt);
            pMax[seg][row] = mt;
            pSum[seg][row] = st;
        }
        __syncthreads();

        if (t < BM) {
            float mt = fmaxf(fmaxf(pMax[0][t], pMax[1][t]),
                             fmaxf(pMax[2][t], pMax[3][t]));
            float st = pSum[0][t] * __expf(pMax[0][t] - mt)
                     + pSum[1][t] * __expf(pMax[1][t] - mt)
                     + pSum[2][t] * __expf(pMax[2][t] - mt)
                     + pSum[3][t] * __expf(pMax[3][t] - mt);

            const float mnew = fmaxf(m_run, mt);
            s_run = s_run * __expf(m_run - mnew) + st * __expf(mt - mnew);
            m_run = mnew;

            const int tg = target[nBase + t];
            if (tg >= vBase && tg < vBase + BN)
                tgt[nBase + t] = Clds[t * LDC + (tg - vBase)];
        }
        // next vt's first in-loop barrier fences Clds/pMax/pSum readers
    }

    if (t < BM) {
        pm[(size_t)blockIdx.y * N_ROWS + nBase + t] = m_run;
        ps[(size_t)blockIdx.y * N_ROWS + nBase + t] = s_run;
    }
}

// -------------------------------------------------------------------------
// Kernel 2: per-row loss from partials + per-block reduction
// grid = 64, block = 128  (64*128 = 8192 rows)
// -------------------------------------------------------------------------
__global__ __launch_bounds__(128)
void ce_rowloss_kernel(const float* __restrict__ pm,
                       const float* __restrict__ ps,
                       const float* __restrict__ tgt,
                       const int*   __restrict__ target,
                       float* __restrict__ bsum,
                       float* __restrict__ bcnt)
{
    __shared__ float sSum[128];
    __shared__ float sCnt[128];

    const int t = threadIdx.x;
    const int n = blockIdx.x * 128 + t;

    float m = -INFINITY, s = 0.0f;
    for (int c = 0; c < VBLOCKS; ++c) {
        const float mi = pm[(size_t)c * N_ROWS + n];
        const float si = ps[(size_t)c * N_ROWS + n];
        const float mn = fmaxf(m, mi);
        s = s * __expf(m - mn) + si * __expf(mi - mn);
        m = mn;
    }
    const int  tg    = target[n];
    const bool valid = (tg != IGNORE_INDEX);
    const float loss = valid ? (__logf(s) + m - tgt[n]) : 0.0f;

    sSum[t] = loss;
    sCnt[t] = valid ? 1.0f : 0.0f;
    __syncthreads();
    for (int off = 64; off > 0; off >>= 1) {
        if (t < off) { sSum[t] += sSum[t + off]; sCnt[t] += sCnt[t + off]; }
        __syncthreads();
    }
    if (t == 0) { bsum[blockIdx.x] = sSum[0]; bcnt[blockIdx.x] = sCnt[0]; }
}

// -------------------------------------------------------------------------
// Kernel 3: final scalar reduce: out = sum(loss) / max(n_valid, 1)
// -------------------------------------------------------------------------
__global__ __launch_bounds__(64)
void ce_final_kernel(const float* __restrict__ bsum,
                     const float* __restrict__ bcnt,
                     float* __restrict__ out)
{
    __shared__ float sSum[64];
    __shared__ float sCnt[64];
    const int t = threadIdx.x;
    sSum[t] = bsum[t];
    sCnt[t] = bcnt[t];
    __syncthreads();
    for (int off = 32; off > 0; off >>= 1) {
        if (t < off) { sSum[t] += sSum[t + off]; sCnt[t] += sCnt[t + off]; }
        __syncthreads();
    }
    if (t == 0) out[0] = sSum[0] / fmaxf(sCnt[0], 1.0f);
}

// -------------------------------------------------------------------------
extern "C" void kernel_launch(void* const* d_in, const int* in_sizes, int n_in,
                              void* d_out, int out_size, void* d_ws, size_t ws_size,
                              hipStream_t stream) {
    const float* x      = (const float*)d_in[0];
    const float* W      = (const float*)d_in[1];
    const int*   target = (const int*)d_in[2];
    float* out = (float*)d_out;

    float* pm   = (float*)d_ws;                        // [VBLOCKS][N]
    float* ps   = pm  + (size_t)VBLOCKS * N_ROWS;      // [VBLOCKS][N]
    float* tgt  = ps  + (size_t)VBLOCKS * N_ROWS;      // [N]
    float* bsum = tgt + N_ROWS;                        // [64]
    float* bcnt = bsum + 64;                           // [64]

    dim3 g1(N_ROWS / BM, VBLOCKS);  // x fastest-varying -> same W slice stays in L2
    ce_partial_kernel<<<g1, 256, 0, stream>>>(x, W, target, pm, ps, tgt);
    ce_rowloss_kernel<<<64, 128, 0, stream>>>(pm, ps, tgt, target, bsum, bcnt);
    ce_final_kernel<<<1, 64, 0, stream>>>(bsum, bcnt, out);
}